// TensorBasis_69587060130233
// MI455X (gfx1250) — compile-verified
//
#include <hip/hip_runtime.h>
#include <hip/hip_bf16.h>

// out[i, p, a*5+b] = sum_{m,n} A[i,m,a] * B[i,n,b] * C[m,n,p]
//   A: (N, 3, 3) f32   B: (N, 5, 5) f32   C: (3, 5, 5) f32   out: (N, 5, 15) f32
//
// Per-atom GEMM on the matrix core:
//   D[ab, p] = sum_mn O[ab, mn] * Cmat[mn, p],  O[ab,mn] = A[m,a]*B[n,b]
// M=ab(15->16), K=mn(15->16), N=p(5->16); 4 chained V_WMMA_F32_16X16X4_F32.
//
// f32 operand layouts (wave32):
//   A 16x4 chunk c: lane<16 -> M=lane, K=4c+{0,1}; lane>=16 -> M=lane-16, K=4c+{2,3}
//   B 4x16 chunk c: lane<16 -> K=4c+{0,1}, N=lane; lane>=16 -> K=4c+{2,3}, N=lane-16
//   C/D 16x16: VGPR v, lane l -> M = v + 8*(l>=16), N = l&15

typedef __attribute__((ext_vector_type(2))) float v2f;
typedef __attribute__((ext_vector_type(8))) float v8f;

#define PF_DIST 16  // atoms ahead to prefetch

__global__ __launch_bounds__(128) void cg_combine_wmma_kernel(
    const float* __restrict__ A,   // (N,3,3)
    const float* __restrict__ B,   // (N,5,5)
    const float* __restrict__ C,   // (3,5,5)
    float* __restrict__ out,       // (N,5,15)
    int n_atoms)
{
    const int lane        = threadIdx.x & 31;
    const int wave_in_blk = threadIdx.x >> 5;
    const int wave_global = blockIdx.x * (blockDim.x >> 5) + wave_in_blk;
    const int total_waves = gridDim.x * (blockDim.x >> 5);

    const bool h   = (lane >> 4) & 1;  // half-wave select
    const int  l16 = lane & 15;

    // ---- fixed per-lane roles ----------------------------------------------
    const int  row  = l16;             // ab index; 15 is padding
    const int  aa   = row / 5;         // 0..3 (3 only for padding row 15)
    const int  bb   = row % 5;
    const int  aaC  = aa > 2 ? 2 : aa; // clamp to avoid OOB load on last atom
    const bool rpad = (row == 15);

    // CG matrix as WMMA B-operand (constant over atoms): K=k (mn), N=l16 (p)
    v2f cb[4];
#pragma unroll
    for (int c = 0; c < 4; ++c) {
#pragma unroll
        for (int v = 0; v < 2; ++v) {
            const int  k    = 4 * c + 2 * (int)h + v;
            const int  m    = k / 5;
            const int  n    = k % 5;
            const bool valC = (k < 15) && (l16 < 5);
            cb[c][v] = valC ? C[m * 25 + n * 5 + l16] : 0.0f;
        }
    }

    // ---- contiguous atom range per wave (sequential HBM streaming) ----------
    const int chunk = (n_atoms + total_waves - 1) / total_waves;
    const int i0    = wave_global * chunk;
    int       i1    = i0 + chunk;
    if (i1 > n_atoms) i1 = n_atoms;

#pragma unroll 2
    for (int i = i0; i < i1; ++i) {
        const float* __restrict__ Ab = A + (size_t)i * 9 + aaC;
        const float* __restrict__ Bb = B + (size_t)i * 25 + bb;

        __builtin_prefetch(A + (size_t)(i + PF_DIST) * 9, 0, 3);
        __builtin_prefetch(B + (size_t)(i + PF_DIST) * 25, 0, 3);

        // 8 distinct loads/lane, immediate offsets -> clause, saddr form
        float a0 = Ab[0];
        float a1 = Ab[3];
        float a2 = Ab[6];
        const float b0 = Bb[0];
        const float b1 = Bb[5];
        const float b2 = Bb[10];
        const float b3 = Bb[15];
        const float b4 = Bb[20];

        if (rpad) { a0 = 0.0f; a1 = 0.0f; a2 = 0.0f; }  // padding row -> 0

        // Half-wave operand selects (v_cndmask), shared across slots
        const float sb_02 = h ? b2   : b0;   // slots (0,0),(1,1)
        const float sb_13 = h ? b3   : b1;   // slot  (0,1)
        const float sb_41 = h ? b1   : b4;   // slots (1,0),(2,1)
        const float sb_30 = h ? b0   : b3;   // slot  (2,0)
        const float sb_24 = h ? b4   : b2;   // slot  (3,0)
        const float sb_3z = h ? 0.0f : b3;   // slot  (3,1), k=15 pad -> 0
        const float sa_01 = h ? a1   : a0;   // slot  (1,0)
        const float sa_12 = h ? a2   : a1;   // slots (2,0),(2,1)

        v2f o[4];
        o[0][0] = a0    * sb_02;
        o[0][1] = a0    * sb_13;
        o[1][0] = sa_01 * sb_41;
        o[1][1] = a1    * sb_02;
        o[2][0] = sa_12 * sb_30;
        o[2][1] = sa_12 * sb_41;
        o[3][0] = a2    * sb_24;
        o[3][1] = a2    * sb_3z;

        v8f acc = {};
#pragma unroll
        for (int c = 0; c < 4; ++c) {
            // 8 args: (neg_a, A, neg_b, B, c_mod, C, reuse_a, reuse_b)
            acc = __builtin_amdgcn_wmma_f32_16x16x4_f32(
                false, o[c], false, cb[c], (short)0, acc, false, false);
        }

        // D: lane(l16)=p column, VGPR v -> ab = v + 8*h.
        if (l16 < 5) {
            float* __restrict__ Oi = out + (size_t)i * 75 + l16 * 15 + 8 * (int)h;
#pragma unroll
            for (int v = 0; v < 8; ++v) {
                if (8 * (int)h + v < 15) {
                    Oi[v] = acc[v];
                }
            }
        }
    }
}

extern "C" void kernel_launch(void* const* d_in, const int* in_sizes, int n_in,
                              void* d_out, int out_size, void* d_ws, size_t ws_size,
                              hipStream_t stream) {
    const float* A = (const float*)d_in[0];   // (N,3,3)
    const float* B = (const float*)d_in[1];   // (N,5,5)
    const float* C = (const float*)d_in[2];   // (3,5,5)
    float* out     = (float*)d_out;           // (N,5,15)

    const int n_atoms = in_sizes[0] / 9;

    const int block = 128;                    // 4 waves / workgroup
    const int waves_per_block = block / 32;
    int blocks = 2048;                        // ~8192 waves, ~74 atoms each
    const int max_blocks = (n_atoms + waves_per_block - 1) / waves_per_block;
    if (blocks > max_blocks) blocks = max_blocks;
    if (blocks < 1) blocks = 1;

    cg_combine_wmma_kernel<<<blocks, block, 0, stream>>>(A, B, C, out, n_atoms);
}